// GNN_12506944766366
// MI455X (gfx1250) — compile-verified
//
#include <hip/hip_runtime.h>
#include <hip/hip_bf16.h>
#include <math.h>

typedef __attribute__((ext_vector_type(2))) float v2f;
typedef __attribute__((ext_vector_type(8))) float v8f;

__device__ __forceinline__ int imin(int a, int b) { return a < b ? a : b; }
__device__ __forceinline__ int imax(int a, int b) { return a > b ? a : b; }

// -----------------------------------------------------------------------------
// Channel-mix GEMM via V_WMMA_F32_16X16X4_F32 (exact f32, matches reference).
// Y[b][m][n] = sum_k A[m][k] * X[b][k][n]
// A: [M x KDIM], X: [B x KDIM x N], Y: [B x M x N]. One wave per 16x16 tile.
// A-frag layout (16x4 f32):  lane L, reg v: A[M = L%16, K = v + 2*(L/16)]
// B-frag layout (4x16 f32):  lane L, reg v: X[K = v + 2*(L/16), N = L%16]
// C/D layout (16x16 f32):    lane L, reg r: D[M = r + 8*(L/16), N = L%16]
// KDIM is a template parameter so the K=3 zero-padding never produces
// divergent (EXEC-predicated) loads and the K=128 path fully unrolls.
// -----------------------------------------------------------------------------
template <int KDIM>
__global__ void gemm_wmma_kernel(const float* __restrict__ A,
                                 const float* __restrict__ X,
                                 float* __restrict__ Y,
                                 int M, int N, int B)
{
    const int ntiles = N >> 4;
    const int tilesPerBatch = (M >> 4) * ntiles;
    const int wave = (int)((blockIdx.x * blockDim.x + threadIdx.x) >> 5);
    const int lane = (int)(threadIdx.x & 31);
    const int b = wave / tilesPerBatch;
    if (b >= B) return;                       // wave-uniform exit (EXEC stays all-1)
    const int t  = wave % tilesPerBatch;
    const int mt = t / ntiles;
    const int nt = t % ntiles;
    const int laneHalf = lane >> 4;           // 0 or 1
    const int laneMod  = lane & 15;

    // Per-lane base pointers for this tile (already offset by laneHalf's K pair).
    const float* __restrict__ ap =
        A + (size_t)(mt * 16 + laneMod) * KDIM + 2 * laneHalf;
    const float* __restrict__ bp =
        X + ((size_t)b * KDIM + 2 * laneHalf) * N + (size_t)(nt * 16 + laneMod);

    v8f acc = {0.f, 0.f, 0.f, 0.f, 0.f, 0.f, 0.f, 0.f};

    if constexpr (KDIM % 4 == 0) {
        // Hot path (K=128): no guards, contiguous A pair -> b64 load,
        // uniform pointer bumps (no per-step 64-bit multiplies).
#pragma unroll
        for (int kk = 0; kk < KDIM / 4; ++kk) {
            v2f afrag, bfrag;
            afrag.x = ap[0];
            afrag.y = ap[1];
            bfrag.x = bp[0];
            bfrag.y = bp[(size_t)N];
            acc = __builtin_amdgcn_wmma_f32_16x16x4_f32(
                false, afrag, false, bfrag, (short)0, acc, false, false);
            ap += 4;
            bp += (size_t)4 * N;
        }
    } else {
        // K=3: one WMMA, zero-pad K=3 slot. laneHalf==0 holds K={0,1},
        // laneHalf==1 holds K={2,pad}. All loads unconditional and in-bounds;
        // padding is a value select (v_cndmask), never an EXEC branch.
        v2f afrag, bfrag;
        afrag.x = ap[0];
        const float aY = ap[laneHalf ? 0 : 1];          // in-bounds for all lanes
        afrag.y = laneHalf ? 0.f : aY;
        bfrag.x = bp[0];
        const float bY = bp[laneHalf ? 0 : (size_t)N];  // in-bounds for all lanes
        bfrag.y = laneHalf ? 0.f : bY;
        acc = __builtin_amdgcn_wmma_f32_16x16x4_f32(
            false, afrag, false, bfrag, (short)0, acc, false, false);
    }

    float* __restrict__ Yb =
        Y + ((size_t)b * M + (size_t)(mt * 16 + 8 * laneHalf)) * N + (size_t)(nt * 16 + laneMod);
#pragma unroll
    for (int r = 0; r < 8; ++r) Yb[(size_t)r * N] = acc[r];
}

// -----------------------------------------------------------------------------
// Affine grid_sample (bilinear, zeros padding, align_corners=False) times the
// analytic BoxPool mask (bilinear sample of a ones image).
// -----------------------------------------------------------------------------
__global__ void sample_kernel(const float* __restrict__ Yin,   // [B,C,H,W]
                              const float* __restrict__ geo,   // [C,2,3]
                              const float* __restrict__ box,   // [C,2,3]
                              float* __restrict__ Out,         // [B,C,H,W]
                              int B, int C, int H, int W)
{
    const size_t total = (size_t)B * C * H * W;
    for (size_t idx = (size_t)blockIdx.x * blockDim.x + threadIdx.x; idx < total;
         idx += (size_t)gridDim.x * blockDim.x) {
        const int w = (int)(idx % W);
        size_t tt = idx / W;
        const int h = (int)(tt % H); tt /= H;
        const int c = (int)(tt % C);
        const int b = (int)(tt / C);

        const float gx = (2.f * w + 1.f) / (float)W - 1.f;
        const float gy = (2.f * h + 1.f) / (float)H - 1.f;

        const float* th = geo + c * 6;
        const float grid_x = th[0] * gx + th[1] * gy + th[2];
        const float grid_y = th[3] * gx + th[4] * gy + th[5];
        const float ix = ((grid_x + 1.f) * (float)W - 1.f) * 0.5f;
        const float iy = ((grid_y + 1.f) * (float)H - 1.f) * 0.5f;
        const float x0f = floorf(ix), y0f = floorf(iy);
        const float wx = ix - x0f, wy = iy - y0f;
        const int x0 = (int)x0f, y0 = (int)y0f;

        const float* __restrict__ img = Yin + ((size_t)b * C + c) * H * W;
        auto tap = [&](int xi, int yi) -> float {
            const bool valid = (xi >= 0) && (xi < W) && (yi >= 0) && (yi < H);
            const int xc = imin(imax(xi, 0), W - 1);
            const int yc = imin(imax(yi, 0), H - 1);
            const float v = img[(size_t)yc * W + xc];
            return valid ? v : 0.f;
        };
        const float samp =
            (1.f - wy) * ((1.f - wx) * tap(x0, y0)     + wx * tap(x0 + 1, y0)) +
                   wy  * ((1.f - wx) * tap(x0, y0 + 1) + wx * tap(x0 + 1, y0 + 1));

        const float* tb = box + c * 6;
        const float bgx = tb[0] * gx + tb[1] * gy + tb[2];
        const float bgy = tb[3] * gx + tb[4] * gy + tb[5];
        const float bix = ((bgx + 1.f) * (float)W - 1.f) * 0.5f;
        const float biy = ((bgy + 1.f) * (float)H - 1.f) * 0.5f;
        const float bx0f = floorf(bix), by0f = floorf(biy);
        const float bwx = bix - bx0f, bwy = biy - by0f;
        const int bx0 = (int)bx0f, by0 = (int)by0f;
        auto ones = [&](int xi, int yi) -> float {
            return ((xi >= 0) && (xi < W) && (yi >= 0) && (yi < H)) ? 1.f : 0.f;
        };
        const float mask =
            (1.f - bwy) * ((1.f - bwx) * ones(bx0, by0)     + bwx * ones(bx0 + 1, by0)) +
                   bwy  * ((1.f - bwx) * ones(bx0, by0 + 1) + bwx * ones(bx0 + 1, by0 + 1));

        Out[idx] = samp * mask;
    }
}

// -----------------------------------------------------------------------------
// MaxPool2d_G (k=2, 64x64 -> 32x32): box-filter conv via LDS integral image,
// first-occurrence argmax, gather the 32x32 window (zeros outside image).
// One block of 256 threads per (b,c) image.
// -----------------------------------------------------------------------------
__global__ void maxpool_g_kernel(const float* __restrict__ In,  // [B*C,64,64]
                                 float* __restrict__ Out)       // [B*C,32,32]
{
    const int H = 64, W = 64;
    __shared__ float img[64 * 64];
    __shared__ float P[65 * 65];          // integral image, P[0][*]=P[*][0]=0
    __shared__ float sv[256];
    __shared__ int   si[256];

    const int bc = (int)blockIdx.x;
    const int tid = (int)threadIdx.x;
    const float* __restrict__ src = In + (size_t)bc * H * W;

    for (int i = tid; i < H * W; i += (int)blockDim.x) img[i] = src[i];
    if (tid < 65) P[tid] = 0.f;           // row 0
    __syncthreads();

    if (tid < 64) {                       // row-prefix for row tid
        float s = 0.f;
        P[(tid + 1) * 65] = 0.f;          // col 0
        for (int q = 0; q < 64; ++q) {
            s += img[tid * 64 + q];
            P[(tid + 1) * 65 + (q + 1)] = s;
        }
    }
    __syncthreads();
    if (tid >= 1 && tid < 65) {           // column accumulate for column tid
        float s = 0.f;
        for (int r = 1; r < 65; ++r) {
            s += P[r * 65 + tid];
            P[r * 65 + tid] = s;
        }
    }
    __syncthreads();

    // conv(p,q) = boxsum rows [p-16,p+15], cols [q-16,q+15] clipped to image.
    float bestV = -3.402823466e+38f;
    int   bestI = 1 << 30;
    for (int k = 0; k < 16; ++k) {
        const int idx = tid + k * 256;    // 0..4095
        const int p = idx >> 6, q = idx & 63;
        const int r0 = imax(p - 16, 0), r1 = imin(p + 15, 63);
        const int c0 = imax(q - 16, 0), c1 = imin(q + 15, 63);
        const float s = P[(r1 + 1) * 65 + (c1 + 1)] - P[r0 * 65 + (c1 + 1)]
                      - P[(r1 + 1) * 65 + c0]       + P[r0 * 65 + c0];
        if (s > bestV || (s == bestV && idx < bestI)) { bestV = s; bestI = idx; }
    }
    sv[tid] = bestV; si[tid] = bestI;
    __syncthreads();
    for (int stride = 128; stride > 0; stride >>= 1) {
        if (tid < stride) {
            const float ov = sv[tid + stride]; const int oi = si[tid + stride];
            if (ov > sv[tid] || (ov == sv[tid] && oi < si[tid])) { sv[tid] = ov; si[tid] = oi; }
        }
        __syncthreads();
    }
    const int argm = si[0];
    const int rm = argm >> 6;             // argm // H (H==W==64)
    const int cm = argm & 63;

    // gathered window = image rows [rm-16, rm+15], cols [cm-16, cm+15], zeros outside
    float* __restrict__ dst = Out + (size_t)bc * 32 * 32;
    for (int o = tid; o < 32 * 32; o += (int)blockDim.x) {
        const int i = o >> 5, j = o & 31;
        const int rr = rm - 16 + i, cc = cm - 16 + j;
        const bool inb = (rr >= 0) && (rr < 64) && (cc >= 0) && (cc < 64);
        dst[o] = inb ? img[rr * 64 + cc] : 0.f;
    }
}

// -----------------------------------------------------------------------------
// Mean pool over 32x32 + dense head. One block per batch element.
// -----------------------------------------------------------------------------
__global__ void head_kernel(const float* __restrict__ feat,  // [B,128,32,32]
                            const float* __restrict__ dw,    // [10,128]
                            const float* __restrict__ db,    // [10]
                            float* __restrict__ out)         // [B,10]
{
    __shared__ float pooled[128];
    const int b = (int)blockIdx.x;
    const int HW = 32 * 32;
    for (int c = (int)threadIdx.x; c < 128; c += (int)blockDim.x) {
        const float* __restrict__ p = feat + ((size_t)b * 128 + c) * HW;
        // mean over W then mean over H (matches reference reduction shape)
        float acc = 0.f;
        for (int r = 0; r < 32; ++r) {
            float rs = 0.f;
            for (int q = 0; q < 32; ++q) rs += p[r * 32 + q];
            acc += rs * (1.f / 32.f);
        }
        pooled[c] = acc * (1.f / 32.f);
    }
    __syncthreads();
    if (threadIdx.x < 10) {
        const int o = (int)threadIdx.x;
        float s = db[o];
        for (int c = 0; c < 128; ++c) s += pooled[c] * dw[o * 128 + c];
        out[b * 10 + o] = s;
    }
}

// -----------------------------------------------------------------------------
// Launch: inLay -> layer0 -> maxPoolG -> layer2 -> layer3 -> head
// -----------------------------------------------------------------------------
extern "C" void kernel_launch(void* const* d_in, const int* in_sizes, int n_in,
                              void* d_out, int out_size, void* d_ws, size_t ws_size,
                              hipStream_t stream) {
    (void)in_sizes; (void)n_in; (void)out_size; (void)ws_size;
    const float* x       = (const float*)d_in[0];  // [8,3,64,64]
    const float* geo0    = (const float*)d_in[1];  // [128,2,3]
    const float* lin0    = (const float*)d_in[2];  // [128,3]
    const float* box0    = (const float*)d_in[3];  // [128,2,3]
    const float* geos    = (const float*)d_in[4];  // [3,128,2,3]
    const float* lins    = (const float*)d_in[5];  // [3,128,128]
    const float* boxes   = (const float*)d_in[6];  // [3,128,2,3]
    const float* dense_w = (const float*)d_in[7];  // [10,128]
    const float* dense_b = (const float*)d_in[8];  // [10]
    float* out = (float*)d_out;                    // [80] ++ feat [8,128,32,32]
    float* feat = out + 80;

    const int B = 8, C = 128;
    float* bufA = (float*)d_ws;                    // 8*128*64*64 f32 = 16 MB
    float* bufB = bufA + (size_t)B * C * 64 * 64;  // second 16 MB

    const dim3 blk(256);
    auto gemmGrid = [&](int M, int N) {            // waves = B*(M/16)*(N/16); 8 waves/block
        return dim3((unsigned)((B * (M / 16) * (N / 16) + 7) / 8));
    };
    auto sampGrid = [&](int H, int W) {
        return dim3((unsigned)(((size_t)B * C * H * W + 255) / 256));
    };

    // inLay: channel mix (K=3) -> sample
    gemm_wmma_kernel<3><<<gemmGrid(C, 4096), blk, 0, stream>>>(lin0, x, bufA, C, 4096, B);
    sample_kernel<<<sampGrid(64, 64), blk, 0, stream>>>(bufA, geo0, box0, bufB, B, C, 64, 64);

    // layer 0
    gemm_wmma_kernel<128><<<gemmGrid(C, 4096), blk, 0, stream>>>(lins, bufB, bufA, C, 4096, B);
    sample_kernel<<<sampGrid(64, 64), blk, 0, stream>>>(bufA, geos, boxes, bufB, B, C, 64, 64);

    // layer 1: MaxPool2d_G  64x64 -> 32x32
    maxpool_g_kernel<<<dim3(B * C), blk, 0, stream>>>(bufB, bufA);

    // layer 2 (H=W=32)
    gemm_wmma_kernel<128><<<gemmGrid(C, 1024), blk, 0, stream>>>(lins + 1 * C * C, bufA, bufB, C, 1024, B);
    sample_kernel<<<sampGrid(32, 32), blk, 0, stream>>>(bufB, geos + 1 * C * 6, boxes + 1 * C * 6,
                                                        bufA, B, C, 32, 32);

    // layer 3 -> feat written straight into d_out
    gemm_wmma_kernel<128><<<gemmGrid(C, 1024), blk, 0, stream>>>(lins + 2 * C * C, bufA, bufB, C, 1024, B);
    sample_kernel<<<sampGrid(32, 32), blk, 0, stream>>>(bufB, geos + 2 * C * 6, boxes + 2 * C * 6,
                                                        feat, B, C, 32, 32);

    // head
    head_kernel<<<dim3(B), blk, 0, stream>>>(feat, dense_w, dense_b, out);
}